// PosEncodingNeRF_73306501808702
// MI455X (gfx1250) — compile-verified
//
#include <hip/hip_runtime.h>
#include <hip/hip_bf16.h>

// PosEncodingNeRF: out[p, 0:4] = c[p, 0:4]
//                  out[p, 4 + 8*i + 2*j + s] = s ? cos(2^i*pi*c[p,j]) : sin(2^i*pi*c[p,j])
// P = 1,048,576 points, OUT_DIM = 100 -> 104,857,600 f32 outputs (~400 MB).
// Store-bandwidth bound: ~18.7 us floor at 23.3 TB/s.
//
// Each thread produces 4 consecutive flat output elements (rows are 100 = 4*25
// floats, so a 4-group never crosses a row): one idx/100, one cached b128 load
// of the point's coords, 4 trans ops (1 per element, optimal), one b128 store.

#define OUT_DIM 100u

__global__ __launch_bounds__(256) void posenc_nerf_kernel(
    const float* __restrict__ coords,   // [P,4] f32
    float* __restrict__ out,            // [P,100] f32 (flat)
    unsigned int nthreads,              // total f32 outputs / 4
    unsigned int P)
{
    unsigned int tid = blockIdx.x * blockDim.x + threadIdx.x;
    if (tid >= nthreads) return;

    unsigned int e0 = tid << 2;              // first flat output element (16B aligned)
    unsigned int p  = e0 / OUT_DIM;          // -> v_mul_hi_u32 + shift
    unsigned int k0 = e0 - p * OUT_DIM;      // in {0,4,...,96}

    // Prefetch the input stream ahead of the store front. Clamp instead of
    // branching: avoids the exec-mask save/cmpx/restore sequence, and the
    // clamped address is always in-bounds (global_prefetch_b8).
    unsigned int ppf = p + 256u;
    ppf = (ppf < P) ? ppf : (P - 1u);
    __builtin_prefetch(coords + 4u * (size_t)ppf, 0, 0);

    const float4 c = *reinterpret_cast<const float4*>(coords + (size_t)p * 4u);

    // Encoding-block decode (values are don't-care when k0 == 0).
    unsigned int e  = (k0 >= 4u) ? (k0 - 4u) : 0u;  // 0..92
    unsigned int fi = e >> 3;                       // frequency index 0..11
    bool hiPair     = (e & 4u) != 0u;               // false: coords {0,1}, true: {2,3}

    // Hardware v_sin_f32(x) = sin(2*pi*x); reference arg is 2^fi * pi * c,
    // so the hardware argument is c * 2^(fi-1): an exact power-of-two scale.
    float scale = __builtin_bit_cast(float, (fi + 126u) << 23);

    float a = (hiPair ? c.z : c.x) * scale;
    float b = (hiPair ? c.w : c.y) * scale;
    // Exact range reduction to [-0.5, 0.5] revolutions (period of v_sin is 1).
    a -= __builtin_rintf(a);
    b -= __builtin_rintf(b);

    float sa = __builtin_amdgcn_sinf(a);
    float ca = __builtin_amdgcn_cosf(a);
    float sb = __builtin_amdgcn_sinf(b);
    float cb = __builtin_amdgcn_cosf(b);

    // Lanes with k0 == 0 (1-2 per wave) write the raw coords instead;
    // select branchlessly (v_cndmask) rather than diverging.
    bool raw = (k0 == 0u);
    float4 r;
    r.x = raw ? c.x : sa;
    r.y = raw ? c.y : ca;
    r.z = raw ? c.z : sb;
    r.w = raw ? c.w : cb;

    *reinterpret_cast<float4*>(out + (size_t)e0) = r;   // coalesced b128 store
}

extern "C" void kernel_launch(void* const* d_in, const int* in_sizes, int n_in,
                              void* d_out, int out_size, void* d_ws, size_t ws_size,
                              hipStream_t stream) {
    const float* coords = (const float*)d_in[0];
    float* out = (float*)d_out;

    unsigned int P        = (unsigned int)(in_sizes[0] / 4);   // points
    unsigned int totalF32 = P * OUT_DIM;                        // 104,857,600
    unsigned int nthreads = totalF32 >> 2;                      // 4 f32 per thread

    dim3 block(256);
    dim3 grid((nthreads + block.x - 1) / block.x);
    posenc_nerf_kernel<<<grid, block, 0, stream>>>(coords, out, nthreads, P);
}